// Decoder_91001767067808
// MI455X (gfx1250) — compile-verified
//
#include <hip/hip_runtime.h>
#include <cstdint>

typedef __attribute__((ext_vector_type(2))) float v2f;
typedef __attribute__((ext_vector_type(8))) float v8f;

#define P_PTS 384   // points per cloud (M == N == 384)
#define HD    256   // offset-head hidden dim
#define HP    257   // padded LDS stride (bank-conflict avoidance)

// ---------------------------------------------------------------------------
// Fused 1x1-conv MLP head:  Y = W2 @ relu(W1 @ X + b1) + b2   (TWO_LAYER)
//                      or   Y = W1 @ X                        (!TWO_LAYER)
// X, Y: (B, C, P) row-major. One workgroup = one (batch, 16-point tile).
// Uses V_WMMA_F32_16X16X4_F32; K stepped by 4 over C.
// A-fragment lane layout (16x4 f32): lane L, vgpr v -> row = L%16, k = 2*(L/16)+v
// B-fragment lane layout (4x16 f32): lane L, vgpr v -> col = L%16, k = 2*(L/16)+v
// C/D (16x16 f32): vgpr r -> row = r + 8*(L/16), col = L%16
// ---------------------------------------------------------------------------
template <int C, bool TWO_LAYER>
__global__ __launch_bounds__(256) void head_kernel(
    const float* __restrict__ X,
    const float* __restrict__ W1, const float* __restrict__ B1,
    const float* __restrict__ W2, const float* __restrict__ B2,
    float* __restrict__ Y)
{
    __shared__ float Xs[16 * C];                    // X tile, [col][C] transposed
    __shared__ float Hs[TWO_LAYER ? 16 * C : 16];   // hidden tile for layer 2

    const int tid  = threadIdx.x;
    const int lane = tid & 31;
    const int wave = tid >> 5;     // 0..7
    const int l16  = lane & 15;
    const int half = lane >> 4;    // 0/1
    const int p0   = blockIdx.x * 16;

    const float* Xb = X + (size_t)blockIdx.y * C * P_PTS;
    float*       Yb = Y + (size_t)blockIdx.y * C * P_PTS;

    // Stage X tile -> LDS transposed ([p][c]); coalesced 64B global segments.
    for (int idx = tid; idx < 16 * C; idx += 256) {
        int c = idx >> 4, p = idx & 15;
        Xs[p * C + c] = Xb[(size_t)c * P_PTS + p0 + p];
    }
    __syncthreads();

    constexpr int RT = C / 16;  // output row-tiles (8 or 16), split across 8 waves

    // ---- layer 1 ----
    for (int rt = wave; rt < RT; rt += 8) {
        const int arow = rt * 16 + l16;  // A-matrix row this lane supplies
        v8f acc = {};
        for (int k = 0; k < C; k += 4) {
            const int kk = k + 2 * half;                       // even -> 8B aligned
            v2f a = *(const v2f*)(W1 + (size_t)arow * C + kk); // weight row slice
            v2f b = *(const v2f*)(&Xs[l16 * C + kk]);          // X column slice
            acc = __builtin_amdgcn_wmma_f32_16x16x4_f32(
                false, a, false, b, (short)0, acc, false, false);
        }
        if constexpr (TWO_LAYER) {
#pragma unroll
            for (int r = 0; r < 8; ++r) {
                int m = rt * 16 + r + 8 * half;
                float v = acc[r] + B1[m];
                Hs[l16 * C + m] = v > 0.f ? v : 0.f;   // relu -> LDS, [col][C]
            }
        } else {
#pragma unroll
            for (int r = 0; r < 8; ++r) {
                int m = rt * 16 + r + 8 * half;
                Yb[(size_t)m * P_PTS + p0 + l16] = acc[r];
            }
        }
    }

    // ---- layer 2 ----
    if constexpr (TWO_LAYER) {
        __syncthreads();
        for (int rt = wave; rt < RT; rt += 8) {
            const int arow = rt * 16 + l16;
            v8f acc = {};
            for (int k = 0; k < C; k += 4) {
                const int kk = k + 2 * half;
                v2f a = *(const v2f*)(W2 + (size_t)arow * C + kk);
                v2f b = *(const v2f*)(&Hs[l16 * C + kk]);
                acc = __builtin_amdgcn_wmma_f32_16x16x4_f32(
                    false, a, false, b, (short)0, acc, false, false);
            }
#pragma unroll
            for (int r = 0; r < 8; ++r) {
                int m = rt * 16 + r + 8 * half;
                Yb[(size_t)m * P_PTS + p0 + l16] = acc[r] + B2[m];
            }
        }
    }
}

// ---------------------------------------------------------------------------
// Pairwise offset residuals. One workgroup = one (b, 16x16 pair tile).
// projXy: (B, H, P) precomputed first-layer projections in workspace.
// ---------------------------------------------------------------------------
__global__ __launch_bounds__(256) void pairwise_kernel(
    const float* __restrict__ pSa,  // w1a @ src_corr
    const float* __restrict__ pSb,  // w1b @ src_corr
    const float* __restrict__ pDa,  // w1a @ dst_corr
    const float* __restrict__ pDb,  // w1b @ dst_corr
    const float* __restrict__ off_b1,
    const float* __restrict__ off_w2,   // (3, H)
    const float* __restrict__ off_b2,   // (3)
    const float* __restrict__ src_xyz,  // (B, 3, M)
    const float* __restrict__ dst_xyz,  // (B, 3, N)
    const uint8_t* __restrict__ src_pad,
    const uint8_t* __restrict__ dst_pad,
    const float* __restrict__ gt_R,     // (B, 3, 3)
    const float* __restrict__ gt_T,     // (B, 3, 1)
    float* __restrict__ out_src, float* __restrict__ out_dst)
{
    __shared__ float sSa[16 * HP], sSb[16 * HP], sDa[16 * HP], sDb[16 * HP];
    __shared__ float sB1[HD];
    __shared__ float sW2[3 * HD];

    const int tid = threadIdx.x;
    const int b   = blockIdx.z;
    const int m0  = blockIdx.y * 16, n0 = blockIdx.x * 16;

    for (int idx = tid; idx < 16 * HD; idx += 256) {
        int h = idx >> 4, p = idx & 15;
        size_t gm = ((size_t)b * HD + h) * P_PTS + m0 + p;
        size_t gn = ((size_t)b * HD + h) * P_PTS + n0 + p;
        sSa[p * HP + h] = pSa[gm];
        sSb[p * HP + h] = pSb[gm];
        sDa[p * HP + h] = pDa[gn];
        sDb[p * HP + h] = pDb[gn];
    }
    if (tid < HD) sB1[tid] = off_b1[tid];
    for (int idx = tid; idx < 3 * HD; idx += 256) sW2[idx] = off_w2[idx];
    __syncthreads();

    const int ml = tid >> 4, nl = tid & 15;
    const int mg = m0 + ml,  ng = n0 + nl;

    float R[9], T[3], sx[3], dx[3];
#pragma unroll
    for (int i = 0; i < 9; ++i) R[i] = gt_R[b * 9 + i];
#pragma unroll
    for (int i = 0; i < 3; ++i) {
        T[i]  = gt_T[b * 3 + i];
        sx[i] = src_xyz[((size_t)b * 3 + i) * P_PTS + mg];
        dx[i] = dst_xyz[((size_t)b * 3 + i) * P_PTS + ng];
    }
    float diff[3];
#pragma unroll
    for (int i = 0; i < 3; ++i) {
        float g = R[i*3+0]*sx[0] + R[i*3+1]*sx[1] + R[i*3+2]*sx[2] + T[i];
        diff[i] = dx[i] - g;                       // dst_gt - src_gt
    }
    float dist = diff[0]*diff[0] + diff[1]*diff[1] + diff[2]*diff[2];
    bool ok = (dist <= 0.25f) && (src_pad[(size_t)b * P_PTS + mg] == 0)
                              && (dst_pad[(size_t)b * P_PTS + ng] == 0);
    float fm = ok ? 1.f : 0.f;

    float as[3] = {0.f, 0.f, 0.f}, ad[3] = {0.f, 0.f, 0.f};
    const float* Sa = &sSa[ml * HP];
    const float* Sb = &sSb[ml * HP];
    const float* Da = &sDa[nl * HP];
    const float* Db = &sDb[nl * HP];
    for (int h = 0; h < HD; ++h) {
        float bb = sB1[h];
        float h1 = fmaxf(Sa[h] + Db[h] + bb, 0.f);   // concat(src, dst) branch
        float h2 = fmaxf(Da[h] + Sb[h] + bb, 0.f);   // concat(dst, src) branch
        float w0 = sW2[h], w1 = sW2[HD + h], w2v = sW2[2 * HD + h];
        as[0] += w0 * h1; as[1] += w1 * h1; as[2] += w2v * h1;
        ad[0] += w0 * h2; ad[1] += w1 * h2; ad[2] += w2v * h2;
    }

    size_t base = (((size_t)b * P_PTS + mg) * P_PTS + ng) * 3;
#pragma unroll
    for (int o = 0; o < 3; ++o) {
        // src_offset_gt = R^T @ diff ; dst_offset_gt = -diff
        float sgt = R[0*3+o]*diff[0] + R[1*3+o]*diff[1] + R[2*3+o]*diff[2];
        out_src[base + o] = (as[o] + off_b2[o] - sgt)     * fm;
        out_dst[base + o] = (ad[o] + off_b2[o] + diff[o]) * fm;
    }
}

// ---------------------------------------------------------------------------
extern "C" void kernel_launch(void* const* d_in, const int* in_sizes, int n_in,
                              void* d_out, int out_size, void* d_ws, size_t ws_size,
                              hipStream_t stream)
{
    (void)in_sizes; (void)n_in; (void)out_size; (void)ws_size;

    const float*   src_fea  = (const float*)d_in[0];
    const float*   dst_fea  = (const float*)d_in[1];
    const float*   src_corr = (const float*)d_in[2];
    const float*   dst_corr = (const float*)d_in[3];
    const float*   src_xyz  = (const float*)d_in[4];
    const float*   dst_xyz  = (const float*)d_in[5];
    const uint8_t* src_pad  = (const uint8_t*)d_in[6];
    const uint8_t* dst_pad  = (const uint8_t*)d_in[7];
    const float*   gt_R     = (const float*)d_in[8];
    const float*   gt_T     = (const float*)d_in[9];
    const float*   cw1 = (const float*)d_in[10];
    const float*   cb1 = (const float*)d_in[11];
    const float*   cw2 = (const float*)d_in[12];
    const float*   cb2 = (const float*)d_in[13];
    const float*   sw1 = (const float*)d_in[14];
    const float*   sb1 = (const float*)d_in[15];
    const float*   sw2 = (const float*)d_in[16];
    const float*   sb2 = (const float*)d_in[17];
    const float*   w1a = (const float*)d_in[18];
    const float*   w1b = (const float*)d_in[19];
    const float*   ob1 = (const float*)d_in[20];
    const float*   ow2 = (const float*)d_in[21];
    const float*   ob2 = (const float*)d_in[22];

    float* out = (float*)d_out;
    float* o_src_pair   = out;                 // (2,256,384) = 196608
    float* o_dst_pair   = out + 196608;
    float* o_src_coarse = out + 393216;        // (2,128,384) =  98304
    float* o_dst_coarse = out + 491520;
    float* o_src_off    = out + 589824;        // (2,384,384,3) = 884736
    float* o_dst_off    = out + 1474560;

    float* ws = (float*)d_ws;                  // 4 x (2,256,384) = 3 MB
    const size_t PR = (size_t)2 * HD * P_PTS;
    float* pSa = ws;
    float* pSb = ws + PR;
    float* pDa = ws + 2 * PR;
    float* pDb = ws + 3 * PR;

    dim3 blk(256);
    dim3 g(P_PTS / 16, 2);   // (point tiles, batch)

    head_kernel<128, true ><<<g, blk, 0, stream>>>(src_fea,  cw1, cb1, cw2, cb2, o_src_coarse);
    head_kernel<128, true ><<<g, blk, 0, stream>>>(dst_fea,  cw1, cb1, cw2, cb2, o_dst_coarse);
    head_kernel<256, true ><<<g, blk, 0, stream>>>(src_corr, sw1, sb1, sw2, sb2, o_src_pair);
    head_kernel<256, true ><<<g, blk, 0, stream>>>(dst_corr, sw1, sb1, sw2, sb2, o_dst_pair);
    head_kernel<256, false><<<g, blk, 0, stream>>>(src_corr, w1a, nullptr, nullptr, nullptr, pSa);
    head_kernel<256, false><<<g, blk, 0, stream>>>(src_corr, w1b, nullptr, nullptr, nullptr, pSb);
    head_kernel<256, false><<<g, blk, 0, stream>>>(dst_corr, w1a, nullptr, nullptr, nullptr, pDa);
    head_kernel<256, false><<<g, blk, 0, stream>>>(dst_corr, w1b, nullptr, nullptr, nullptr, pDb);

    dim3 gp(P_PTS / 16, P_PTS / 16, 2);
    pairwise_kernel<<<gp, blk, 0, stream>>>(pSa, pSb, pDa, pDb, ob1, ow2, ob2,
                                            src_xyz, dst_xyz, src_pad, dst_pad,
                                            gt_R, gt_T, o_src_off, o_dst_off);
}